// CameraFrequency_70368744178157
// MI455X (gfx1250) — compile-verified
//
#include <hip/hip_runtime.h>
#include <math.h>

// Problem constants (from reference): feats [B=4, N=16, S=4096, FD=128],
// matrix [B=4, C=8, 4, 4], S = C*P with P=512, FD = F*4 with F=32.
// out[b,n,c,p,f,i] = sum_j (Lf diag(exp(Df)) Uf @ matrix[b,c])[i,j] * feats[b,n,c,p,f,j]

typedef __attribute__((ext_vector_type(2))) float v2f;
typedef __attribute__((ext_vector_type(8))) float v8f;

__global__ __launch_bounds__(256) void camfreq_wmma_kernel(
    const float* __restrict__ feats,
    const float* __restrict__ matrix,
    const float* __restrict__ Lp,
    const float* __restrict__ Dp,
    const float* __restrict__ Up,
    float* __restrict__ out)
{
    const int lane = threadIdx.x & 31;
    const int wv   = threadIdx.x >> 5;
    const int W    = blockIdx.x * 8 + wv;          // 16384 waves total

    const int g  = W & 7;          // frequency group: 4 freqs = 16 columns
    const int tq = (W >> 3) & 3;   // quarter of the 32 p-tiles (8 tiles each)
    const int c  = (W >> 5) & 7;
    const int n  = (W >> 8) & 15;
    const int b  = W >> 12;

    const int row16 = lane & 15;   // A-operand row / tile row for loads
    const int khalf = lane >> 4;   // this lane supplies k = 2*khalf, 2*khalf+1
    const int ncol  = lane & 15;   // D/B column within tile
    const int fo    = ncol >> 2;   // which freq (0..3) of the group owns this column
    const int oi    = ncol & 3;    // output dim i within that freq

    // ---------------- per-wave one-time: build 4 B operands ----------------
    // freq-matrix M_f = L diag(exp(D)) U, then M5 = M_f @ matrix[b,c].
    // B_f(k, 4*fo+i) = delta(f,fo) * M5[i][k]. Lane supplies M5[oi][2*khalf + {0,1}].
    const int f = 4 * g + fo;
    const float l0 = Lp[f*6+0], l1 = Lp[f*6+1], l2 = Lp[f*6+2];
    const float l3 = Lp[f*6+3], l4 = Lp[f*6+4], l5 = Lp[f*6+5];
    const float u0 = Up[f*6+0], u1 = Up[f*6+1], u2 = Up[f*6+2];
    const float u3 = Up[f*6+3], u4 = Up[f*6+4], u5 = Up[f*6+5];
    const float e0 = expf(Dp[f*4+0]), e1 = expf(Dp[f*4+1]);
    const float e2 = expf(Dp[f*4+2]), e3 = expf(Dp[f*4+3]);

    // rows of freq = L @ diag(e) @ U
    const float r00 = e0,    r01 = e0*u0,         r02 = e0*u1,                 r03 = e0*u2;
    const float r10 = l0*e0, r11 = l0*e0*u0 + e1, r12 = l0*e0*u1 + e1*u3,      r13 = l0*e0*u2 + e1*u4;
    const float r20 = l1*e0, r21 = l1*e0*u0 + l2*e1,
                r22 = l1*e0*u1 + l2*e1*u3 + e2,
                r23 = l1*e0*u2 + l2*e1*u4 + e2*u5;
    const float r30 = l3*e0, r31 = l3*e0*u0 + l4*e1,
                r32 = l3*e0*u1 + l4*e1*u3 + l5*e2,
                r33 = l3*e0*u2 + l4*e1*u4 + l5*e2*u5 + e3;

    // select row oi of freq
    const float fr0 = (oi==0) ? r00 : (oi==1) ? r10 : (oi==2) ? r20 : r30;
    const float fr1 = (oi==0) ? r01 : (oi==1) ? r11 : (oi==2) ? r21 : r31;
    const float fr2 = (oi==0) ? r02 : (oi==1) ? r12 : (oi==2) ? r22 : r32;
    const float fr3 = (oi==0) ? r03 : (oi==1) ? r13 : (oi==2) ? r23 : r33;

    // M5[oi][k] = sum_j freq[oi][j] * matrix[b][c][j][k], for k = 2*khalf + {0,1}
    const float* mat = matrix + (b*8 + c) * 16;
    const int col0 = 2 * khalf;
    const float mv0 = fr0*mat[0*4+col0]   + fr1*mat[1*4+col0]   + fr2*mat[2*4+col0]   + fr3*mat[3*4+col0];
    const float mv1 = fr0*mat[0*4+col0+1] + fr1*mat[1*4+col0+1] + fr2*mat[2*4+col0+1] + fr3*mat[3*4+col0+1];

    const v2f zero2 = {0.0f, 0.0f};
    const v2f bown  = {mv0, mv1};
    const v2f bop0 = (fo == 0) ? bown : zero2;
    const v2f bop1 = (fo == 1) ? bown : zero2;
    const v2f bop2 = (fo == 2) ? bown : zero2;
    const v2f bop3 = (fo == 3) ? bown : zero2;

    // ---------------- streaming loop over 8 p-tiles of 16 rows ----------------
    const size_t base = ((size_t)(b*16 + n) * 4096 + (size_t)c * 512) * 128;
    const float* fbase = feats + base + (size_t)row16 * 128 + 16*g + 2*khalf;
    float*       obase = out   + base + 16*g + ncol;

    for (int t = tq * 8; t < tq * 8 + 8; ++t) {
        const float* fp = fbase + (size_t)t * (16 * 128);
        // A_f operands: lane row16/row16+16 holds row row16, k = {0,1}/{2,3}
        const v2f a0 = *(const v2f*)(fp + 0);
        const v2f a1 = *(const v2f*)(fp + 4);
        const v2f a2 = *(const v2f*)(fp + 8);
        const v2f a3 = *(const v2f*)(fp + 12);

        v8f acc = {};
        acc = __builtin_amdgcn_wmma_f32_16x16x4_f32(false, a0, false, bop0, (short)0, acc, false, false);
        acc = __builtin_amdgcn_wmma_f32_16x16x4_f32(false, a1, false, bop1, (short)0, acc, false, false);
        acc = __builtin_amdgcn_wmma_f32_16x16x4_f32(false, a2, false, bop2, (short)0, acc, false, false);
        acc = __builtin_amdgcn_wmma_f32_16x16x4_f32(false, a3, false, bop3, (short)0, acc, false, false);

        // D layout: VGPR r -> row (t*16 + r + 8*khalf), col = 16*g + ncol
        float* op = obase + (size_t)(t * 16 + 8 * khalf) * 128;
#pragma unroll
        for (int r = 0; r < 8; ++r) {
            op[(size_t)r * 128] = acc[r];
        }
    }
}

extern "C" void kernel_launch(void* const* d_in, const int* in_sizes, int n_in,
                              void* d_out, int out_size, void* d_ws, size_t ws_size,
                              hipStream_t stream) {
    (void)in_sizes; (void)n_in; (void)out_size; (void)d_ws; (void)ws_size;
    const float* feats  = (const float*)d_in[0];
    const float* matrix = (const float*)d_in[1];
    const float* Lp     = (const float*)d_in[2];
    const float* Dp     = (const float*)d_in[3];
    const float* Up     = (const float*)d_in[4];
    float* out = (float*)d_out;

    // 4 (b) * 16 (n) * 8 (c) * 8 (g) * 4 (tile-quarter) = 16384 waves
    // = 2048 blocks of 256 threads (8 wave32 per block).
    camfreq_wmma_kernel<<<2048, 256, 0, stream>>>(feats, matrix, Lp, Dp, Up, out);
}